// GraphSAGE_20263655702631
// MI455X (gfx1250) — compile-verified
//
#include <hip/hip_runtime.h>
#include <hip/hip_bf16.h>
#include <stdint.h>

typedef float v2f __attribute__((ext_vector_type(2)));
typedef float v8f __attribute__((ext_vector_type(8)));

#define EPS 1e-5f

// ---------------------------------------------------------------- utilities
__global__ void fill0_kernel(float* __restrict__ p, long long n) {
    long long t = (long long)blockIdx.x * blockDim.x + threadIdx.x;
    if (t < n) p[t] = 0.0f;
}

// deg[dst] += 1 per edge
__global__ void deg_kernel(const long long* __restrict__ dst, float* __restrict__ deg, int E) {
    int e = blockIdx.x * blockDim.x + threadIdx.x;
    if (e < E) atomicAdd(&deg[dst[e]], 1.0f);
}

// deg -> 1/max(deg,1)
__global__ void invdeg_kernel(float* __restrict__ deg, int N) {
    int i = blockIdx.x * blockDim.x + threadIdx.x;
    if (i < N) deg[i] = 1.0f / fmaxf(deg[i], 1.0f);
}

// agg[dst,:] += h[src,:]  (4-feature chunks per thread, float4 fast path)
__global__ void scatter_kernel(const float* __restrict__ h,
                               const long long* __restrict__ src,
                               const long long* __restrict__ dst,
                               float* __restrict__ agg,
                               int E, int d, int nchunk) {
    long long t = (long long)blockIdx.x * blockDim.x + threadIdx.x;
    long long total = (long long)E * nchunk;
    if (t >= total) return;
    int e = (int)(t / nchunk);
    int c = (int)(t % nchunk) * 4;
    long long s  = src[e];
    long long dd = dst[e];
    const float* hp = h   + s  * (long long)d + c;
    float*       ap = agg + dd * (long long)d + c;
    int m = d - c; if (m > 4) m = 4;
    if (m == 4) {
        float4 v = *reinterpret_cast<const float4*>(hp);
        atomicAdd(ap + 0, v.x);
        atomicAdd(ap + 1, v.y);
        atomicAdd(ap + 2, v.z);
        atomicAdd(ap + 3, v.w);
    } else {
        for (int i = 0; i < m; ++i) atomicAdd(ap + i, hp[i]);
    }
}

// ---------------------------------------------------------------- WMMA transform
// out[16x16 tile] = (agg*invdeg) @ Wl + h @ Wr + bl   via V_WMMA_F32_16X16X4_F32
// one wave per 16x16 output tile; N % 16 == 0 and DOUT % 16 == 0 by construction.
// Compile-time DIN/DOUT => fully unrolled, guard-free, vectorized fragment loads.
template <int DIN, int DOUT>
__global__ void sage_transform_kernel(const float* __restrict__ agg,
                                      const float* __restrict__ h,
                                      const float* __restrict__ invdeg,
                                      const float* __restrict__ Wl,
                                      const float* __restrict__ Wr,
                                      const float* __restrict__ bl,
                                      float* __restrict__ out,
                                      int N) {
    const int wave = threadIdx.x >> 5;
    const int lane = threadIdx.x & 31;
    const int colTiles = DOUT >> 4;
    const int rowTiles = N >> 4;
    const int totalTiles = rowTiles * colTiles;
    const int tile = blockIdx.x * (blockDim.x >> 5) + wave;
    if (tile >= totalTiles) return;            // uniform per wave -> EXEC all-ones below
    const int rowTile = tile / colTiles;
    const int colTile = tile % colTiles;
    const int l    = lane & 15;
    const int half = lane >> 4;                // 0: K=k0..k0+1, 1: K=k0+2..k0+3
    const int row  = rowTile * 16 + l;         // A-matrix row M for this lane
    const int col  = colTile * 16 + l;         // B/C/D column N for this lane

    v8f c = {0.f, 0.f, 0.f, 0.f, 0.f, 0.f, 0.f, 0.f};
    const float idg = invdeg[row];
    const float* aggRow = agg + (long long)row * DIN;
    const float* hRow   = h   + (long long)row * DIN;

    if constexpr (DIN >= 4) {
        // pass 1: (agg * invdeg) @ Wl   -- ka = k0+2*half, ka+1 <= DIN-1 always
        #pragma unroll
        for (int k0 = 0; k0 < DIN; k0 += 4) {
            const int ka = k0 + 2 * half;
            v2f av = *reinterpret_cast<const v2f*>(aggRow + ka);   // 8B-aligned b64 load
            v2f a; a.x = av.x * idg; a.y = av.y * idg;
            v2f b;
            b.x = Wl[(ka + 0) * DOUT + col];
            b.y = Wl[(ka + 1) * DOUT + col];
            c = __builtin_amdgcn_wmma_f32_16x16x4_f32(false, a, false, b,
                                                      (short)0, c, false, false);
        }
        // pass 2: h @ Wr
        #pragma unroll
        for (int k0 = 0; k0 < DIN; k0 += 4) {
            const int ka = k0 + 2 * half;
            v2f a = *reinterpret_cast<const v2f*>(hRow + ka);
            v2f b;
            b.x = Wr[(ka + 0) * DOUT + col];
            b.y = Wr[(ka + 1) * DOUT + col];
            c = __builtin_amdgcn_wmma_f32_16x16x4_f32(false, a, false, b,
                                                      (short)0, c, false, false);
        }
    } else {
        // DIN == 2: lanes with half==1 would need K=2,3 (nonexistent) -> mask to 0,
        // branch-free so EXEC stays all-ones.
        const float m = half ? 0.0f : 1.0f;
        {
            v2f av = *reinterpret_cast<const v2f*>(aggRow);        // K=0,1
            v2f a; a.x = av.x * idg * m; a.y = av.y * idg * m;
            v2f b;
            b.x = Wl[0 * DOUT + col] * m;
            b.y = Wl[1 * DOUT + col] * m;
            c = __builtin_amdgcn_wmma_f32_16x16x4_f32(false, a, false, b,
                                                      (short)0, c, false, false);
        }
        {
            v2f hv = *reinterpret_cast<const v2f*>(hRow);
            v2f a; a.x = hv.x * m; a.y = hv.y * m;
            v2f b;
            b.x = Wr[0 * DOUT + col] * m;
            b.y = Wr[1 * DOUT + col] * m;
            c = __builtin_amdgcn_wmma_f32_16x16x4_f32(false, a, false, b,
                                                      (short)0, c, false, false);
        }
    }

    // bias + store: C/D layout: VGPR v, lanes0-15 -> M=v, lanes16-31 -> M=v+8, N=l
    const float bias = bl[col];
    #pragma unroll
    for (int v = 0; v < 8; ++v) {
        int r = rowTile * 16 + v + 8 * half;
        out[(long long)r * DOUT + col] = c[v] + bias;
    }
}

// ---------------------------------------------------------------- batchnorm
// per-column sums / sum-of-squares; blockDim.x == d
__global__ void bn_stats_kernel(const float* __restrict__ h,
                                float* __restrict__ sums, float* __restrict__ sumsq,
                                int N, int d) {
    int col = threadIdx.x;
    float s = 0.f, q = 0.f;
    for (int r = blockIdx.x; r < N; r += gridDim.x) {
        float v = h[(long long)r * d + col];
        s += v;
        q += v * v;
    }
    atomicAdd(&sums[col], s);
    atomicAdd(&sumsq[col], q);
}

__global__ void bn_relu_kernel(float* __restrict__ h,
                               const float* __restrict__ sums,
                               const float* __restrict__ sumsq,
                               const float* __restrict__ gamma,
                               const float* __restrict__ beta,
                               int N, int d) {
    long long t = (long long)blockIdx.x * blockDim.x + threadIdx.x;
    if (t >= (long long)N * d) return;
    int col = (int)(t % d);
    float invN = 1.0f / (float)N;
    float mean = sums[col] * invN;
    float var  = sumsq[col] * invN - mean * mean;
    float v = (h[t] - mean) * rsqrtf(var + EPS) * gamma[col] + beta[col];
    h[t] = v > 0.0f ? v : 0.0f;
}

// ---------------------------------------------------------------- pooling + head
__global__ void pool_sum_kernel(const float* __restrict__ h,
                                const long long* __restrict__ batch,
                                float* __restrict__ pooled, int N, int d) {
    long long t = (long long)blockIdx.x * blockDim.x + threadIdx.x;
    if (t >= (long long)N * d) return;
    int r = (int)(t / d);
    int c = (int)(t % d);
    atomicAdd(&pooled[batch[r] * (long long)d + c], h[t]);
}

__global__ void pool_cnt_kernel(const long long* __restrict__ batch,
                                float* __restrict__ cnt, int N) {
    int i = blockIdx.x * blockDim.x + threadIdx.x;
    if (i < N) atomicAdd(&cnt[batch[i]], 1.0f);
}

// pooled[g,:]/cnt[g] @ Wf1 + bf1 -> relu ; grid = G blocks, block = dh threads
__global__ void fc1_kernel(const float* __restrict__ pooled, const float* __restrict__ cnt,
                           const float* __restrict__ Wf1, const float* __restrict__ bf1,
                           float* __restrict__ o, int d, int dh) {
    int g = blockIdx.x;
    int t = threadIdx.x;
    float inv = 1.0f / fmaxf(cnt[g], 1.0f);
    float s = bf1[t];
    for (int k = 0; k < d; ++k)
        s += pooled[(long long)g * d + k] * inv * Wf1[(long long)k * dh + t];
    o[g * dh + t] = fmaxf(s, 0.0f);
}

__global__ void fc2_kernel(const float* __restrict__ o, const float* __restrict__ Wf2,
                           const float* __restrict__ bf2, float* __restrict__ out,
                           int G, int dh) {
    int g = blockIdx.x * blockDim.x + threadIdx.x;
    if (g >= G) return;
    float s = bf2[0];
    for (int k = 0; k < dh; ++k) s += o[g * dh + k] * Wf2[k];
    out[g] = s;
}

// ---------------------------------------------------------------- host launch
static inline long long cdivll(long long a, long long b) { return (a + b - 1) / b; }

struct LayerBufs {
    float *aggbuf, *invdeg, *sums, *sumsq;
    const long long *src, *dst;
    int N, E;
    hipStream_t stream;
};

template <int DIN, int DOUT>
static void run_layer(const LayerBufs& L, const float* hin,
                      const float* Wl, const float* bl, const float* Wr,
                      const float* g, const float* be, float* hout) {
    const int T = 256;
    const int nchunk = (DIN + 3) / 4;
    fill0_kernel<<<cdivll((long long)L.N * DIN, T), T, 0, L.stream>>>(
        L.aggbuf, (long long)L.N * DIN);
    scatter_kernel<<<cdivll((long long)L.E * nchunk, T), T, 0, L.stream>>>(
        hin, L.src, L.dst, L.aggbuf, L.E, DIN, nchunk);
    const int tiles = (L.N / 16) * (DOUT / 16);
    const int wavesPerBlock = 8;               // 256 threads = 8 waves
    sage_transform_kernel<DIN, DOUT>
        <<<cdivll(tiles, wavesPerBlock), wavesPerBlock * 32, 0, L.stream>>>(
            L.aggbuf, hin, L.invdeg, Wl, Wr, bl, hout, L.N);
    fill0_kernel<<<1, 256, 0, L.stream>>>(L.sums, 128);
    fill0_kernel<<<1, 256, 0, L.stream>>>(L.sumsq, 128);
    bn_stats_kernel<<<512, DOUT, 0, L.stream>>>(hout, L.sums, L.sumsq, L.N, DOUT);
    bn_relu_kernel<<<cdivll((long long)L.N * DOUT, T), T, 0, L.stream>>>(
        hout, L.sums, L.sumsq, g, be, L.N, DOUT);
}

extern "C" void kernel_launch(void* const* d_in, const int* in_sizes, int n_in,
                              void* d_out, int out_size, void* d_ws, size_t ws_size,
                              hipStream_t stream) {
    (void)n_in; (void)ws_size; (void)out_size;
    const float*     x    = (const float*)d_in[0];
    const long long* ei   = (const long long*)d_in[1];
    const long long* batch= (const long long*)d_in[2];
    const float *Wl1=(const float*)d_in[3],  *bl1=(const float*)d_in[4],  *Wr1=(const float*)d_in[5];
    const float *g1 =(const float*)d_in[6],  *be1=(const float*)d_in[7];
    const float *Wl2=(const float*)d_in[8],  *bl2=(const float*)d_in[9],  *Wr2=(const float*)d_in[10];
    const float *g2 =(const float*)d_in[11], *be2=(const float*)d_in[12];
    const float *Wl3=(const float*)d_in[13], *bl3=(const float*)d_in[14], *Wr3=(const float*)d_in[15];
    const float *g3 =(const float*)d_in[16], *be3=(const float*)d_in[17];
    const float *Wf1=(const float*)d_in[18], *bf1=(const float*)d_in[19];
    const float *Wf2=(const float*)d_in[20], *bf2=(const float*)d_in[21];
    float* out = (float*)d_out;

    const int N = in_sizes[0] / 2;       // x is [N, 2]
    const int E = in_sizes[1] / 2;       // edge_index is [2, E]
    const int G = 64;
    const long long* src = ei;           // row 0
    const long long* dst = ei + E;       // row 1

    // ---- workspace carve
    char* ws = (char*)d_ws;
    size_t off = 0;
    auto carve = [&](size_t bytes) -> char* {
        char* p = ws + off;
        off += (bytes + 255) & ~(size_t)255;
        return p;
    };
    float* invdeg = (float*)carve((size_t)N * 4);          // deg then invdeg
    float* aggbuf = (float*)carve((size_t)N * 64 * 4);     // max din = 64
    float* bufA   = (float*)carve((size_t)N * 128 * 4);    // h1 (32) then h3 (128)
    float* bufB   = (float*)carve((size_t)N * 64 * 4);     // h2 (64)
    float* sums   = (float*)carve(128 * 4);
    float* sumsq  = (float*)carve(128 * 4);
    float* pooled = (float*)carve((size_t)G * 128 * 4);
    float* cnt    = (float*)carve((size_t)G * 4);
    float* fc1buf = (float*)carve((size_t)G * 64 * 4);

    const int T = 256;

    // ---- degree -> invdeg (shared across layers)
    fill0_kernel<<<cdivll(N, T), T, 0, stream>>>(invdeg, N);
    deg_kernel<<<cdivll(E, T), T, 0, stream>>>(dst, invdeg, E);
    invdeg_kernel<<<cdivll(N, T), T, 0, stream>>>(invdeg, N);

    LayerBufs L{aggbuf, invdeg, sums, sumsq, src, dst, N, E, stream};
    run_layer<2, 32>  (L, x,    Wl1, bl1, Wr1, g1, be1, bufA);
    run_layer<32, 64> (L, bufA, Wl2, bl2, Wr2, g2, be2, bufB);
    run_layer<64, 128>(L, bufB, Wl3, bl3, Wr3, g3, be3, bufA);

    // ---- global mean pool + MLP head
    fill0_kernel<<<cdivll((long long)G * 128, T), T, 0, stream>>>(pooled, (long long)G * 128);
    fill0_kernel<<<1, 64, 0, stream>>>(cnt, G);
    pool_sum_kernel<<<cdivll((long long)N * 128, T), T, 0, stream>>>(bufA, batch, pooled, N, 128);
    pool_cnt_kernel<<<cdivll(N, T), T, 0, stream>>>(batch, cnt, N);
    fc1_kernel<<<G, 64, 0, stream>>>(pooled, cnt, Wf1, bf1, fc1buf, 128, 64);
    fc2_kernel<<<1, G, 0, stream>>>(fc1buf, Wf2, bf2, out, G, 64);
}